// GetCostVolume_3410204033793
// MI455X (gfx1250) — compile-verified
//
#include <hip/hip_runtime.h>
#include <cstdint>

#define CCH   32
#define HH_   128
#define WW_   256
#define NDISP 48
#define DCHUNK 8

#if defined(__HIP_DEVICE_COMPILE__) && __has_builtin(__builtin_amdgcn_global_load_async_to_lds_b128)
#define HAVE_ASYNC_LDS 1
#else
#define HAVE_ASYNC_LDS 0
#endif

typedef int v4i __attribute__((ext_vector_type(4)));
typedef __attribute__((address_space(1))) v4i* gv4i_p;   // global (device) int4*
typedef __attribute__((address_space(3))) v4i* lv4i_p;   // LDS int4*

__global__ __launch_bounds__(256) void cost_volume_kernel(
    const float* __restrict__ x,
    const float* __restrict__ y,
    const float* __restrict__ disp,
    float* __restrict__ out)
{
    // Two y-rows (all 32 channels) staged in LDS: 2*32*256 f32 = 64 KB
    __shared__ float sY[2 * CCH * WW_];

    const int hh  = blockIdx.x;            // 0..127
    const int d0  = blockIdx.y * DCHUNK;   // disparity chunk base
    const int tid = threadIdx.x;           // 0..255
    const int ww  = tid;

    // ---- vertical coordinate (uniform per block), reference FP order ----
    const float gy  = (float)hh / 63.5f - 1.0f;               // (H-1)/2 = 63.5
    const float iyv = ((gy + 1.0f) * 128.0f - 1.0f) * 0.5f;
    const float y0f = floorf(iyv);
    const float wy  = iyv - y0f;
    const int   y0i = (int)y0f;
    const int   y1i = y0i + 1;

    // ---- stage two y rows into LDS; out-of-range rows become zeros ------
    // 2 rows * 32 ch * 256 f32 = 4096 float4; 256 threads -> 16 float4 each
    for (int i = tid; i < 2 * CCH * (WW_ / 4); i += 256) {
        const int r   = i >> 11;           // / 2048
        const int rem = i & 2047;
        const int c   = rem >> 6;          // / 64
        const int w4  = rem & 63;
        const int yr  = (r == 0) ? y0i : y1i;
        float* dstf = &sY[(r * CCH + c) * WW_ + (w4 << 2)];
        if (yr >= 0 && yr < HH_) {
            const float* srcf = &y[((size_t)(c * HH_ + yr)) * WW_ + (w4 << 2)];
#if HAVE_ASYNC_LDS
            __builtin_amdgcn_global_load_async_to_lds_b128(
                (gv4i_p)srcf, (lv4i_p)dstf, 0, 0);
#else
            *(float4*)dstf = *(const float4*)srcf;
#endif
        } else {
            *(float4*)dstf = make_float4(0.f, 0.f, 0.f, 0.f);
        }
    }

#if HAVE_ASYNC_LDS
#if __has_builtin(__builtin_amdgcn_s_wait_asynccnt)
    __builtin_amdgcn_s_wait_asynccnt(0);
#else
    asm volatile("s_wait_asynccnt 0" ::: "memory");
#endif
#endif
    __syncthreads();

    // ---- left features for this (hh, ww) held in registers --------------
    float regX[CCH];
#pragma unroll
    for (int c = 0; c < CCH; ++c)
        regX[c] = x[((size_t)(c * HH_ + hh)) * WW_ + ww];

    const size_t HW      = (size_t)HH_ * WW_;
    const size_t DHW     = (size_t)NDISP * HW;
    const size_t base_hw = (size_t)hh * WW_ + ww;

    const float omwy = 1.0f - wy;

    for (int dd = 0; dd < DCHUNK; ++dd) {
        const int d = d0 + dd;
        if (dd + 1 < DCHUNK)
            __builtin_prefetch(&disp[((size_t)(d + 1)) * HW + base_hw], 0, 0);

        const float dv    = disp[(size_t)d * HW + base_hw];
        // reference FP order: cur_x/((w-1)/2) - 1, then ((gx+1)*w - 1)*0.5
        const float cur_x = (float)ww - dv;
        const float gx    = cur_x / 127.5f - 1.0f;
        const float ixv   = ((gx + 1.0f) * 256.0f - 1.0f) * 0.5f;
        const float x0f   = floorf(ixv);
        const float wx    = ixv - x0f;
        const int   x0    = (int)x0f;
        const int   x1    = x0 + 1;

        const float m0 = (x0 >= 0 && x0 < WW_) ? 1.0f : 0.0f;
        const float m1 = (x1 >= 0 && x1 < WW_) ? 1.0f : 0.0f;
        const int x0c = min(max(x0, 0), WW_ - 1);
        const int x1c = min(max(x1, 0), WW_ - 1);

        // y-mask already folded into zeroed LDS rows; fold x-mask into weights
        const float omwx = 1.0f - wx;
        const float a00 = omwx * omwy * m0;
        const float a01 = wx   * omwy * m1;
        const float a10 = omwx * wy   * m0;
        const float a11 = wx   * wy   * m1;

        float* outL = out + (size_t)d * HW + base_hw;                     // ch 0..C-1
        float* outR = out + (size_t)CCH * DHW + (size_t)d * HW + base_hw; // ch C..2C-1

#pragma unroll
        for (int c = 0; c < CCH; ++c) {
            const float v00 = sY[c * WW_ + x0c];
            const float v01 = sY[c * WW_ + x1c];
            const float v10 = sY[(CCH + c) * WW_ + x0c];
            const float v11 = sY[(CCH + c) * WW_ + x1c];
            const float res = v00 * a00 + v01 * a01 + v10 * a10 + v11 * a11;
            outL[(size_t)c * DHW] = regX[c];
            outR[(size_t)c * DHW] = res;
        }
    }
}

extern "C" void kernel_launch(void* const* d_in, const int* in_sizes, int n_in,
                              void* d_out, int out_size, void* d_ws, size_t ws_size,
                              hipStream_t stream) {
    const float* x    = (const float*)d_in[0];
    const float* y    = (const float*)d_in[1];
    const float* disp = (const float*)d_in[2];
    float* out        = (float*)d_out;

    dim3 grid(HH_, NDISP / DCHUNK);   // (128, 6) blocks
    cost_volume_kernel<<<grid, 256, 0, stream>>>(x, y, disp, out);
}